// TransformerBlock_71150428225747
// MI455X (gfx1250) — compile-verified
//
#include <hip/hip_runtime.h>
#include <math.h>

#define HEADS 4
#define CH 16
#define DMODEL 64

typedef __attribute__((ext_vector_type(2))) float v2f;
typedef __attribute__((ext_vector_type(8))) float v8f;

// ---------------------------------------------------------------------------
// init: amax = -inf, denom = 0   (workspace is not re-poisoned between calls)
// ---------------------------------------------------------------------------
__global__ void tc_init_kernel(float* __restrict__ amax, float* __restrict__ denom, int n)
{
    int i = blockIdx.x * blockDim.x + threadIdx.x;
    if (i < n) {
        amax[i]  = -INFINITY;
        denom[i] = 0.0f;
    }
}

// ---------------------------------------------------------------------------
// fused q/k/v/skip GEMM:  [N,64] x [64,64] (+bias) for 4 weight matrices.
// One block = one 16-node tile, 8 waves, 16 wave-tasks = {matrix 0..3} x
// {16-col tile 0..3}; each task accumulates with V_WMMA_F32_16X16X4_F32.
// ---------------------------------------------------------------------------
__device__ __forceinline__ const float* sel4(int m, const float* a, const float* b,
                                             const float* c, const float* d)
{
    return m == 0 ? a : (m == 1 ? b : (m == 2 ? c : d));
}
__device__ __forceinline__ float* sel4o(int m, float* a, float* b, float* c, float* d)
{
    return m == 0 ? a : (m == 1 ? b : (m == 2 ? c : d));
}

__global__ void __launch_bounds__(256)
tc_qkvs_gemm_kernel(const float* __restrict__ x,
                    const float* __restrict__ Wq, const float* __restrict__ bq,
                    const float* __restrict__ Wk, const float* __restrict__ bk,
                    const float* __restrict__ Wv, const float* __restrict__ bv,
                    const float* __restrict__ Ws, const float* __restrict__ bs,
                    float* __restrict__ qo, float* __restrict__ ko,
                    float* __restrict__ vo, float* __restrict__ out, int N)
{
    const int lane = threadIdx.x & 31;
    const int wave = threadIdx.x >> 5;
    const int hh   = lane >> 4;        // half-wave: selects K pair / row group
    const int r    = lane & 15;

    const int nodeBase = blockIdx.x * 16;

    // two tasks per wave: wave and wave+8
    const int t0 = wave,      t1 = wave + 8;
    const int m0 = t0 >> 2,   m1 = t1 >> 2;          // which weight matrix
    const int c0 = (t0 & 3) * 16, c1 = (t1 & 3) * 16; // output column tile

    const float* W0 = sel4(m0, Wq, Wk, Wv, Ws);
    const float* W1 = sel4(m1, Wq, Wk, Wv, Ws);
    const float* B0 = sel4(m0, bq, bk, bv, bs);
    const float* B1 = sel4(m1, bq, bk, bv, bs);
    float* O0 = sel4o(m0, qo, ko, vo, out);
    float* O1 = sel4o(m1, qo, ko, vo, out);

    int arow = nodeBase + r;
    if (arow >= N) arow = N - 1;
    const float* xrow = x + (size_t)arow * DMODEL;

    v8f acc0 = {};
    v8f acc1 = {};

    #pragma unroll
    for (int kk = 0; kk < DMODEL; kk += 4) {
        // A 16x4 f32 frag: lanes 0-15 hold K={kk,kk+1}, lanes 16-31 K={kk+2,kk+3}
        v2f a;
        a[0] = xrow[kk + 2 * hh + 0];
        a[1] = xrow[kk + 2 * hh + 1];
        // B 4x16 f32 frags: VGPR0 rows K={kk | kk+2}, VGPR1 rows K={kk+1 | kk+3}
        v2f b0, b1;
        b0[0] = W0[(size_t)(kk + 2 * hh + 0) * DMODEL + c0 + r];
        b0[1] = W0[(size_t)(kk + 2 * hh + 1) * DMODEL + c0 + r];
        b1[0] = W1[(size_t)(kk + 2 * hh + 0) * DMODEL + c1 + r];
        b1[1] = W1[(size_t)(kk + 2 * hh + 1) * DMODEL + c1 + r];

        acc0 = __builtin_amdgcn_wmma_f32_16x16x4_f32(false, a, false, b0,
                                                     (short)0, acc0, false, false);
        acc1 = __builtin_amdgcn_wmma_f32_16x16x4_f32(false, a, false, b1,
                                                     (short)0, acc1, false, false);
    }

    const float bias0 = B0[c0 + r];
    const float bias1 = B1[c1 + r];
    #pragma unroll
    for (int i = 0; i < 8; ++i) {
        const int row = nodeBase + i + 8 * hh;   // C/D layout: VGPR i -> M=i / M=i+8
        if (row < N) {
            O0[(size_t)row * DMODEL + c0 + r] = acc0[i] + bias0;
            O1[(size_t)row * DMODEL + c1 + r] = acc1[i] + bias1;
        }
    }
}

// ---------------------------------------------------------------------------
// float atomic max via integer atomics (valid with -inf init, mixed signs)
// ---------------------------------------------------------------------------
__device__ __forceinline__ void atomicMaxF32(float* addr, float val)
{
    if (val >= 0.0f) {
        atomicMax((int*)addr, __float_as_int(val));
    } else {
        atomicMin((unsigned int*)addr, __float_as_uint(val));
    }
}

// ---------------------------------------------------------------------------
// pass B: alpha_raw = <q[dst,h,:], k[src,h,:]> / sqrt(C); segment max by dst
// ---------------------------------------------------------------------------
__global__ void __launch_bounds__(256)
tc_edge_alpha_kernel(const int* __restrict__ ei,
                     const float* __restrict__ q, const float* __restrict__ k,
                     float* __restrict__ alpha, float* __restrict__ amax, int E)
{
    int id = blockIdx.x * blockDim.x + threadIdx.x;
    if (id >= E * HEADS) return;
    const int e = id >> 2, h = id & 3;
    const int src = ei[e];
    const int dst = ei[E + e];

    const float4* qp = (const float4*)(q + (size_t)dst * DMODEL + h * CH);
    const float4* kp = (const float4*)(k + (size_t)src * DMODEL + h * CH);
    float s = 0.0f;
    #pragma unroll
    for (int i = 0; i < 4; ++i) {
        float4 a = qp[i], b = kp[i];
        s += a.x * b.x + a.y * b.y + a.z * b.z + a.w * b.w;
    }
    s *= 0.25f;   // 1/sqrt(16)
    alpha[id] = s;
    atomicMaxF32(amax + (size_t)dst * HEADS + h, s);
}

// ---------------------------------------------------------------------------
// pass C: ealpha = exp(alpha - amax[dst]); segment sum -> denom
// ---------------------------------------------------------------------------
__global__ void __launch_bounds__(256)
tc_edge_exp_kernel(const int* __restrict__ ei, float* __restrict__ alpha,
                   const float* __restrict__ amax, float* __restrict__ denom, int E)
{
    int id = blockIdx.x * blockDim.x + threadIdx.x;
    if (id >= E * HEADS) return;
    const int e = id >> 2, h = id & 3;
    const int dst = ei[E + e];
    const float ea = __expf(alpha[id] - amax[(size_t)dst * HEADS + h]);
    alpha[id] = ea;
    atomicAdd(denom + (size_t)dst * HEADS + h, ea);
}

// ---------------------------------------------------------------------------
// pass D: out[dst] += v[src] * (ealpha / (denom[dst] + 1e-16))
// ---------------------------------------------------------------------------
__global__ void __launch_bounds__(256)
tc_edge_agg_kernel(const int* __restrict__ ei, const float* __restrict__ alpha,
                   const float* __restrict__ denom, const float* __restrict__ v,
                   float* __restrict__ out, int E)
{
    int id = blockIdx.x * blockDim.x + threadIdx.x;
    if (id >= E * HEADS) return;
    const int e = id >> 2, h = id & 3;
    const int src = ei[e];
    const int dst = ei[E + e];
    const float a = alpha[id] / (denom[(size_t)dst * HEADS + h] + 1e-16f);

    const float* vp = v + (size_t)src * DMODEL + h * CH;
    float* op = out + (size_t)dst * DMODEL + h * CH;
    #pragma unroll
    for (int c = 0; c < CH; ++c) {
        atomicAdd(op + c, vp[c] * a);
    }
}

// ---------------------------------------------------------------------------
extern "C" void kernel_launch(void* const* d_in, const int* in_sizes, int n_in,
                              void* d_out, int out_size, void* d_ws, size_t ws_size,
                              hipStream_t stream)
{
    const float* x  = (const float*)d_in[0];
    const int*   ei = (const int*)d_in[1];
    const float* Wq = (const float*)d_in[2];
    const float* bq = (const float*)d_in[3];
    const float* Wk = (const float*)d_in[4];
    const float* bk = (const float*)d_in[5];
    const float* Wv = (const float*)d_in[6];
    const float* bv = (const float*)d_in[7];
    const float* Ws = (const float*)d_in[8];
    const float* bs = (const float*)d_in[9];
    float* out = (float*)d_out;

    const int N = in_sizes[0] / DMODEL;
    const int E = in_sizes[1] / 2;

    // workspace layout (floats)
    float* ws    = (float*)d_ws;
    float* qb    = ws;                              // N*64
    float* kb    = qb + (size_t)N * DMODEL;         // N*64
    float* vb    = kb + (size_t)N * DMODEL;         // N*64
    float* alpha = vb + (size_t)N * DMODEL;         // E*4
    float* amax  = alpha + (size_t)E * HEADS;       // N*4
    float* denom = amax + (size_t)N * HEADS;        // N*4

    {
        int n = N * HEADS;
        tc_init_kernel<<<(n + 255) / 256, 256, 0, stream>>>(amax, denom, n);
    }
    {
        int blocks = (N + 15) / 16;
        tc_qkvs_gemm_kernel<<<blocks, 256, 0, stream>>>(
            x, Wq, bq, Wk, bk, Wv, bv, Ws, bs, qb, kb, vb, out, N);
    }
    {
        int n = E * HEADS;
        int blocks = (n + 255) / 256;
        tc_edge_alpha_kernel<<<blocks, 256, 0, stream>>>(ei, qb, kb, alpha, amax, E);
        tc_edge_exp_kernel<<<blocks, 256, 0, stream>>>(ei, alpha, amax, denom, E);
        tc_edge_agg_kernel<<<blocks, 256, 0, stream>>>(ei, alpha, denom, vb, out, E);
    }
}